// Net_22239340658905
// MI455X (gfx1250) — compile-verified
//
#include <hip/hip_runtime.h>
#include <math.h>

#define N_NODES   100000
#define N_EDGES   1600000
#define N_FEAT    500
#define HIDDEN    64
#define N_CLASSES 16
#define M_TILES   (N_NODES / 16)   // 6250, exact

typedef __attribute__((ext_vector_type(2))) float v2f;
typedef __attribute__((ext_vector_type(8))) float v8f;

// ---------------- degree / norm ----------------

__global__ void k_init_deg(float* deg) {
    int i = blockIdx.x * blockDim.x + threadIdx.x;
    if (i < N_NODES) deg[i] = 1.0f;   // self-loop weight
}

__global__ void k_accum_deg(const int* __restrict__ col,
                            const float* __restrict__ w,
                            float* __restrict__ deg) {
    int e = blockIdx.x * blockDim.x + threadIdx.x;
    if (e < N_EDGES) atomicAdd(&deg[col[e]], w[e]);
}

__global__ void k_dinv(const float* __restrict__ deg, float* __restrict__ dinv) {
    int i = blockIdx.x * blockDim.x + threadIdx.x;
    if (i < N_NODES) {
        float d = deg[i];
        dinv[i] = (d > 0.0f) ? rsqrtf(d) : 0.0f;
    }
}

__global__ void k_norm(const int* __restrict__ row, const int* __restrict__ col,
                       const float* __restrict__ w, const float* __restrict__ dinv,
                       float* __restrict__ norm) {
    int e = blockIdx.x * blockDim.x + threadIdx.x;
    if (e < N_EDGES) norm[e] = dinv[row[e]] * w[e] * dinv[col[e]];
}

// ---------------- GEMM1: h1 = X @ W1 (100000x500 @ 500x64), fp32 WMMA ----------------
// One wave per 16-row M tile; 4 accumulators cover all 64 output columns so X
// is streamed from HBM exactly once.

__global__ __launch_bounds__(128) void k_gemm1(const float* __restrict__ x,
                                               const float* __restrict__ W1,
                                               float* __restrict__ h1) {
    const int wave  = threadIdx.x >> 5;
    const int lane  = threadIdx.x & 31;
    const int mtile = blockIdx.x * 4 + wave;
    if (mtile >= M_TILES) return;                  // wave-uniform: EXEC stays full
    const int row0 = mtile * 16;
    const int half = lane >> 4;                    // 0: lanes 0-15 (K 0,1), 1: lanes 16-31 (K 2,3)
    const int l16  = lane & 15;

    const float* arow = x + (size_t)(row0 + l16) * N_FEAT + half * 2;
    v8f c0 = {}, c1 = {}, c2 = {}, c3 = {};

    for (int k = 0; k < N_FEAT; k += 4) {
        v2f a;
        a.x = arow[k];
        a.y = arow[k + 1];
        const float* bp = W1 + (size_t)(k + half * 2) * HIDDEN + l16;
        v2f b0, b1, b2, b3;
        b0.x = bp[0];   b0.y = bp[HIDDEN + 0];
        b1.x = bp[16];  b1.y = bp[HIDDEN + 16];
        b2.x = bp[32];  b2.y = bp[HIDDEN + 32];
        b3.x = bp[48];  b3.y = bp[HIDDEN + 48];
        c0 = __builtin_amdgcn_wmma_f32_16x16x4_f32(false, a, false, b0, (short)0, c0, false, false);
        c1 = __builtin_amdgcn_wmma_f32_16x16x4_f32(false, a, false, b1, (short)0, c1, false, false);
        c2 = __builtin_amdgcn_wmma_f32_16x16x4_f32(false, a, false, b2, (short)0, c2, false, false);
        c3 = __builtin_amdgcn_wmma_f32_16x16x4_f32(false, a, false, b3, (short)0, c3, false, false);
    }

    // C/D layout: VGPR r -> M = r (lanes 0-15) / r+8 (lanes 16-31), N = lane%16
    float* out = h1 + (size_t)(row0 + half * 8) * HIDDEN + l16;
    for (int r = 0; r < 8; ++r) {
        out[(size_t)r * HIDDEN + 0]  = c0[r];
        out[(size_t)r * HIDDEN + 16] = c1[r];
        out[(size_t)r * HIDDEN + 32] = c2[r];
        out[(size_t)r * HIDDEN + 48] = c3[r];
    }
}

// ---------------- aggregation, layer 1 ----------------
// agg1[i,f] = b1[f] + dinv[i]^2 * h1[i,f]   (self-loop contribution + bias)
__global__ void k_init_agg1(const float* __restrict__ h1,
                            const float* __restrict__ dinv,
                            const float* __restrict__ b1,
                            float* __restrict__ agg1) {
    int t = blockIdx.x * blockDim.x + threadIdx.x;
    if (t < N_NODES * HIDDEN) {
        int i = t / HIDDEN, f = t % HIDDEN;
        float s = dinv[i];
        agg1[t] = b1[f] + s * s * h1[t];
    }
}

__global__ void k_scatter1(const int* __restrict__ row, const int* __restrict__ col,
                           const float* __restrict__ norm,
                           const float* __restrict__ h1,
                           float* __restrict__ agg1) {
    unsigned t = blockIdx.x * blockDim.x + threadIdx.x;
    if (t < (unsigned)N_EDGES * HIDDEN) {
        unsigned e = t / HIDDEN, f = t % HIDDEN;
        float v = norm[e] * h1[(size_t)row[e] * HIDDEN + f];
        atomicAdd(&agg1[(size_t)col[e] * HIDDEN + f], v);
    }
}

// ---------------- GEMM2: h2 = relu(agg1) @ W2 (100000x64 @ 64x16), fp32 WMMA ----------------

__global__ __launch_bounds__(128) void k_gemm2(const float* __restrict__ agg1,
                                               const float* __restrict__ W2,
                                               float* __restrict__ h2) {
    const int wave  = threadIdx.x >> 5;
    const int lane  = threadIdx.x & 31;
    const int mtile = blockIdx.x * 4 + wave;
    if (mtile >= M_TILES) return;
    const int row0 = mtile * 16;
    const int half = lane >> 4;
    const int l16  = lane & 15;

    const float* arow = agg1 + (size_t)(row0 + l16) * HIDDEN + half * 2;
    v8f c = {};

    for (int k = 0; k < HIDDEN; k += 4) {
        v2f a;
        a.x = fmaxf(arow[k], 0.0f);        // fused ReLU
        a.y = fmaxf(arow[k + 1], 0.0f);
        const float* bp = W2 + (size_t)(k + half * 2) * N_CLASSES + l16;
        v2f b;
        b.x = bp[0];
        b.y = bp[N_CLASSES];
        c = __builtin_amdgcn_wmma_f32_16x16x4_f32(false, a, false, b, (short)0, c, false, false);
    }

    float* out = h2 + (size_t)(row0 + half * 8) * N_CLASSES + l16;
    for (int r = 0; r < 8; ++r)
        out[(size_t)r * N_CLASSES] = c[r];
}

// ---------------- aggregation, layer 2 ----------------

__global__ void k_init_agg2(const float* __restrict__ h2,
                            const float* __restrict__ dinv,
                            const float* __restrict__ b2,
                            float* __restrict__ agg2) {
    int t = blockIdx.x * blockDim.x + threadIdx.x;
    if (t < N_NODES * N_CLASSES) {
        int i = t / N_CLASSES, cc = t % N_CLASSES;
        float s = dinv[i];
        agg2[t] = b2[cc] + s * s * h2[t];
    }
}

__global__ void k_scatter2(const int* __restrict__ row, const int* __restrict__ col,
                           const float* __restrict__ norm,
                           const float* __restrict__ h2,
                           float* __restrict__ agg2) {
    unsigned t = blockIdx.x * blockDim.x + threadIdx.x;
    if (t < (unsigned)N_EDGES * N_CLASSES) {
        unsigned e = t / N_CLASSES, cc = t % N_CLASSES;
        float v = norm[e] * h2[(size_t)row[e] * N_CLASSES + cc];
        atomicAdd(&agg2[(size_t)col[e] * N_CLASSES + cc], v);
    }
}

// ---------------- log-softmax over 16 classes ----------------

__global__ void k_logsoftmax(const float* __restrict__ agg2, float* __restrict__ out) {
    int i = blockIdx.x * blockDim.x + threadIdx.x;
    if (i >= N_NODES) return;
    const float* v = agg2 + (size_t)i * N_CLASSES;
    float m = v[0];
    #pragma unroll
    for (int c = 1; c < N_CLASSES; ++c) m = fmaxf(m, v[c]);
    float s = 0.0f;
    #pragma unroll
    for (int c = 0; c < N_CLASSES; ++c) s += expf(v[c] - m);
    float l = logf(s);
    float* o = out + (size_t)i * N_CLASSES;
    #pragma unroll
    for (int c = 0; c < N_CLASSES; ++c) o[c] = v[c] - m - l;
}

// ---------------- launcher ----------------

extern "C" void kernel_launch(void* const* d_in, const int* in_sizes, int n_in,
                              void* d_out, int out_size, void* d_ws, size_t ws_size,
                              hipStream_t stream) {
    const float* x  = (const float*)d_in[0];
    const int*   ei = (const int*)d_in[1];          // [2, E] flat
    const float* ew = (const float*)d_in[2];
    const float* W1 = (const float*)d_in[3];
    const float* b1 = (const float*)d_in[4];
    const float* W2 = (const float*)d_in[5];
    const float* b2 = (const float*)d_in[6];
    float* out = (float*)d_out;

    const int* row = ei;
    const int* col = ei + N_EDGES;

    float* ws   = (float*)d_ws;
    float* deg  = ws;                                   // 100000
    float* dinv = deg  + N_NODES;                       // 100000
    float* norm = dinv + N_NODES;                       // 1.6M
    float* h1   = norm + N_EDGES;                       // 6.4M
    float* agg1 = h1   + (size_t)N_NODES * HIDDEN;      // 6.4M
    float* h2   = agg1 + (size_t)N_NODES * HIDDEN;      // 1.6M
    float* agg2 = h2   + (size_t)N_NODES * N_CLASSES;   // 1.6M

    const int B = 256;

    k_init_deg <<<(N_NODES + B - 1) / B, B, 0, stream>>>(deg);
    k_accum_deg<<<(N_EDGES + B - 1) / B, B, 0, stream>>>(col, ew, deg);
    k_dinv     <<<(N_NODES + B - 1) / B, B, 0, stream>>>(deg, dinv);
    k_norm     <<<(N_EDGES + B - 1) / B, B, 0, stream>>>(row, col, ew, dinv, norm);

    k_gemm1<<<(M_TILES + 3) / 4, 128, 0, stream>>>(x, W1, h1);

    k_init_agg1<<<(N_NODES * HIDDEN + B - 1) / B, B, 0, stream>>>(h1, dinv, b1, agg1);
    {
        unsigned total = (unsigned)N_EDGES * HIDDEN;
        k_scatter1<<<(total + B - 1) / B, B, 0, stream>>>(row, col, norm, h1, agg1);
    }

    k_gemm2<<<(M_TILES + 3) / 4, 128, 0, stream>>>(agg1, W2, h2);

    k_init_agg2<<<(N_NODES * N_CLASSES + B - 1) / B, B, 0, stream>>>(h2, dinv, b2, agg2);
    {
        unsigned total = (unsigned)N_EDGES * N_CLASSES;
        k_scatter2<<<(total + B - 1) / B, B, 0, stream>>>(row, col, norm, h2, agg2);
    }

    k_logsoftmax<<<(N_NODES + B - 1) / B, B, 0, stream>>>(agg2, out);
}